// Quantization_82617990906038
// MI455X (gfx1250) — compile-verified
//
#include <hip/hip_runtime.h>

typedef __attribute__((ext_vector_type(2))) float v2f;
typedef __attribute__((ext_vector_type(8))) float v8f;

#define B_   32
#define D_   64
#define H_   64
#define W_   64
#define K_   512
#define NPIX (B_*H_*W_)                          /* 131072 */
#define M_TILE 16
#define NBLOCKS (NPIX / M_TILE)                  /* 8192 */

#define QUANT_OFF ((size_t)0)
#define ENC_OFF   ((size_t)(B_*D_*H_*W_))        /* 8388608 */
#define DIST_OFF  (ENC_OFF + (size_t)NPIX)       /* 8519680 */
#define LOSS_OFF  (DIST_OFF + (size_t)NPIX * K_) /* 75628544 */

#define XS_LD 66
#define ES_LD 66

// ---------------------------------------------------------------------------
// Precompute ||e_k||^2 for all 512 codebook rows into d_ws[0..511].
// ---------------------------------------------------------------------------
__global__ void vq_enorm_kernel(const float* __restrict__ emb,
                                float* __restrict__ enorm) {
    int c = blockIdx.x * blockDim.x + threadIdx.x;
    if (c < K_) {
        float s = 0.f;
        #pragma unroll 8
        for (int d = 0; d < D_; ++d) {
            float v = emb[c * D_ + d];
            s += v * v;
        }
        enorm[c] = s;
    }
}

// ---------------------------------------------------------------------------
// Main kernel: one block = 16 pixels x 512 codes.
//   - distances via f32 WMMA 16x16x4 (16 chained ops => K=64)
//   - argmin (encoding), gather (quantized_st), partial MSE loss
// ---------------------------------------------------------------------------
__global__ __launch_bounds__(256, 2)
void vq_main_kernel(const float* __restrict__ x,
                    const float* __restrict__ emb,
                    const float* __restrict__ enorm_g,
                    float* __restrict__ out,
                    float* __restrict__ blockLoss) {
    __shared__ float xs[M_TILE * XS_LD];      // 16 pixels x 64 ch (padded)
    __shared__ float es[128 * ES_LD];         // 128 codes x 64 ch (padded)
    __shared__ float enorm_s[K_];
    __shared__ float xnorm_s[M_TILE];
    __shared__ float candD[8][M_TILE];
    __shared__ int   candI[8][M_TILE];
    __shared__ int   finalIdx[M_TILE];
    __shared__ float wloss[8];

    const int tid  = threadIdx.x;
    const int wave = tid >> 5;
    const int lane = tid & 31;
    const int half = lane >> 4;     // 0: K=0,1 slots; 1: K=2,3 slots
    const int l16  = lane & 15;

    const int pixelBase = blockIdx.x * M_TILE;
    const int b  = pixelBase >> 12;           // / (H*W)
    const int hw = pixelBase & 4095;
    const int h  = hw >> 6;
    const int w0 = hw & 63;                   // 16 | 64, tile never crosses a row
    const size_t xbase = (size_t)b * (D_ * H_ * W_) + (size_t)h * W_ + w0;

    // ---- stage x tile: xs[p][d] = x[b, d, h, w0+p] (coalesced along w) ----
    #pragma unroll
    for (int rep = 0; rep < 4; ++rep) {
        int linear = tid + rep * 256;         // 0..1023
        int d = linear >> 4;
        int p = linear & 15;
        xs[p * XS_LD + d] = x[xbase + (size_t)d * (H_ * W_) + p];
    }
    // ---- stage codebook norms ----
    enorm_s[tid]       = enorm_g[tid];
    enorm_s[tid + 256] = enorm_g[tid + 256];
    __syncthreads();

    // ---- per-pixel ||x||^2 ----
    if (tid < M_TILE) {
        float s = 0.f;
        #pragma unroll 8
        for (int d = 0; d < D_; ++d) {
            float v = xs[tid * XS_LD + d];
            s += v * v;
        }
        xnorm_s[tid] = s;
    }

    float bestD[8];
    int   bestI[8];
    #pragma unroll
    for (int r = 0; r < 8; ++r) { bestD[r] = 3.4e38f; bestI[r] = 0; }

    for (int pass = 0; pass < 4; ++pass) {
        __syncthreads();   // previous pass finished reading es (pass 0: staging done)
        // ---- stage 128-code tile: es[c][d] (coalesced: 64 threads / code row) ----
        #pragma unroll
        for (int j = 0; j < 32; ++j) {
            int linear = tid + j * 256;       // 0..8191
            int c = linear >> 6;
            int d = linear & 63;
            es[c * ES_LD + d] = emb[(size_t)(pass * 128 + c) * D_ + d];
        }
        __syncthreads();

        // ---- wave computes its 16(pixels) x 16(codes) dot tile, K=64 ----
        v8f acc = {};
        const int   crow = wave * 16 + l16;                     // code row in es
        const float* xrow = &xs[l16 * XS_LD + half * 2];
        const float* erow = &es[crow * ES_LD + half * 2];
        #pragma unroll
        for (int ks = 0; ks < 16; ++ks) {
            v2f a, bf;
            a.x  = xrow[4 * ks];     a.y  = xrow[4 * ks + 1];
            bf.x = erow[4 * ks];     bf.y = erow[4 * ks + 1];
            acc = __builtin_amdgcn_wmma_f32_16x16x4_f32(
                false, a, false, bf, (short)0, acc, false, false);
        }

        // ---- epilogue: distances + running argmin ----
        const int codeBase = pass * 128 + wave * 16;
        const int code = codeBase + l16;
        #pragma unroll
        for (int r = 0; r < 8; ++r) {
            int m = r + half * 8;   // C/D layout: lanes 16-31 hold rows M=r+8
            float dist = xnorm_s[m] + enorm_s[code] - 2.0f * acc[r];
            out[DIST_OFF + (size_t)(pixelBase + m) * K_ + code] = dist;
            if (dist < bestD[r] || (dist == bestD[r] && code < bestI[r])) {
                bestD[r] = dist;
                bestI[r] = code;
            }
        }
    }

    // ---- argmin reduce across the 16 lanes of each half-wave ----
    #pragma unroll
    for (int r = 0; r < 8; ++r) {
        float d0 = bestD[r];
        int   i0 = bestI[r];
        #pragma unroll
        for (int off = 8; off >= 1; off >>= 1) {
            float dOth = __shfl_xor(d0, off, 32);
            int   iOth = __shfl_xor(i0, off, 32);
            if (dOth < d0 || (dOth == d0 && iOth < i0)) { d0 = dOth; i0 = iOth; }
        }
        if (l16 == 0) {
            candD[wave][r + half * 8] = d0;
            candI[wave][r + half * 8] = i0;
        }
    }
    __syncthreads();

    // ---- merge the 8 wave candidates per pixel (fixed order, tie -> low idx) ----
    if (tid < M_TILE) {
        float bd = candD[0][tid];
        int   bi = candI[0][tid];
        #pragma unroll
        for (int w = 1; w < 8; ++w) {
            float dd = candD[w][tid];
            int   ii = candI[w][tid];
            if (dd < bd || (dd == bd && ii < bi)) { bd = dd; bi = ii; }
        }
        finalIdx[tid] = bi;
        out[ENC_OFF + pixelBase + tid] = (float)bi;
    }
    __syncthreads();

    // ---- quantized_st (== gathered codebook rows) + loss partial ----
    float lacc = 0.f;
    #pragma unroll
    for (int rep = 0; rep < 4; ++rep) {
        int linear = tid + rep * 256;
        int d = linear >> 4;
        int p = linear & 15;
        int idx = finalIdx[p];
        float q  = emb[(size_t)idx * D_ + d];
        float xv = xs[p * XS_LD + d];
        float diff = q - xv;
        lacc += diff * diff;
        out[QUANT_OFF + xbase + (size_t)d * (H_ * W_) + p] = q;
    }
    #pragma unroll
    for (int off = 16; off >= 1; off >>= 1)
        lacc += __shfl_xor(lacc, off, 32);
    if (lane == 0) wloss[wave] = lacc;
    __syncthreads();
    if (tid == 0) {
        float s = 0.f;
        #pragma unroll
        for (int w = 0; w < 8; ++w) s += wloss[w];
        blockLoss[blockIdx.x] = s;   // deterministic per-block partial
    }
}

// ---------------------------------------------------------------------------
// Deterministic final reduction of 8192 block partials -> loss scalar.
// ---------------------------------------------------------------------------
__global__ void vq_loss_kernel(const float* __restrict__ blockLoss,
                               float* __restrict__ out) {
    __shared__ float s[256];
    float a = 0.f;
    #pragma unroll 4
    for (int j = 0; j < 32; ++j)
        a += blockLoss[threadIdx.x + j * 256];
    s[threadIdx.x] = a;
    __syncthreads();
    for (int stride = 128; stride >= 1; stride >>= 1) {
        if (threadIdx.x < stride) s[threadIdx.x] += s[threadIdx.x + stride];
        __syncthreads();
    }
    if (threadIdx.x == 0)
        out[LOSS_OFF] = s[0] * (1.0f / (float)(B_ * D_ * H_ * W_));
}

// ---------------------------------------------------------------------------
extern "C" void kernel_launch(void* const* d_in, const int* in_sizes, int n_in,
                              void* d_out, int out_size, void* d_ws, size_t ws_size,
                              hipStream_t stream) {
    (void)in_sizes; (void)n_in; (void)out_size; (void)ws_size;
    const float* x   = (const float*)d_in[0];   // [32,64,64,64] f32
    const float* emb = (const float*)d_in[1];   // [512,64] f32
    float* out = (float*)d_out;

    float* enorm     = (float*)d_ws;            // 512 floats
    float* blockLoss = enorm + K_;              // 8192 floats

    vq_enorm_kernel<<<2, 256, 0, stream>>>(emb, enorm);
    vq_main_kernel<<<NBLOCKS, 256, 0, stream>>>(x, emb, enorm, out, blockLoss);
    vq_loss_kernel<<<1, 256, 0, stream>>>(blockLoss, out);
}